// VecRippleCarryAdder108Bit_43860206027324
// MI455X (gfx1250) — compile-verified
//
#include <hip/hip_runtime.h>
#include <stdint.h>

// 108-bit ripple-carry adder over {0,1} float inputs.
// wave32 ballot packs each 108-float row into a 108-bit integer held in SGPRs,
// one 128-bit add-with-carry replaces the whole soft-logic carry chain, then
// each lane unpacks its own nibble and stores one float4 (b128 NT store).
// Streaming input uses CDNA5 async global->LDS DMA (ASYNCcnt), double-buffered.

#define BITS            108
#define ROW_BYTES       (BITS * 4)            // 432 B per row
#define WAVES_PER_BLOCK 8
#define BLOCK_THREADS   (WAVES_PER_BLOCK * 32)
#define BUF_FLOATS      128                   // 108 floats padded to 512B

typedef float v4f __attribute__((ext_vector_type(4)));

__device__ __forceinline__ uint32_t lds_addr32(const void* p) {
  // Generic pointers to __shared__ carry the LDS byte offset in the low 32 bits
  // (flat-shared aperture).
  return (uint32_t)(uintptr_t)p;
}

__device__ __forceinline__ void async_copy_b128(uint32_t lds_addr, uint64_t gaddr) {
  // CDNA5 async DMA: global memory -> LDS, tracked by ASYNCcnt.
  asm volatile("global_load_async_to_lds_b128 %0, %1, off"
               :: "v"(lds_addr), "v"(gaddr)
               : "memory");
}

__device__ __forceinline__ void s_wait_async_le2() {
  asm volatile("s_wait_asynccnt 0x2" ::: "memory");
}
__device__ __forceinline__ void s_wait_async_le0() {
  asm volatile("s_wait_asynccnt 0x0" ::: "memory");
}

__global__ __launch_bounds__(BLOCK_THREADS)
void VecRippleCarryAdder108Bit_kernel(const float* __restrict__ A,
                                      const float* __restrict__ B,
                                      float* __restrict__ S,
                                      float* __restrict__ C,
                                      int nrows) {
  __shared__ float lsA[WAVES_PER_BLOCK][2][BUF_FLOATS];
  __shared__ float lsB[WAVES_PER_BLOCK][2][BUF_FLOATS];

  const int  lane   = (int)(threadIdx.x & 31u);
  const int  wave   = (int)(threadIdx.x >> 5);
  const long gwave  = (long)blockIdx.x * WAVES_PER_BLOCK + wave;
  const long stride = (long)gridDim.x * WAVES_PER_BLOCK;

  // Per-lane LDS destinations (27 lanes x 16B cover one 432B row).
  uint32_t aOff[2], bOff[2];
  aOff[0] = lds_addr32(&lsA[wave][0][0]) + (uint32_t)lane * 16u;
  aOff[1] = lds_addr32(&lsA[wave][1][0]) + (uint32_t)lane * 16u;
  bOff[0] = lds_addr32(&lsB[wave][0][0]) + (uint32_t)lane * 16u;
  bOff[1] = lds_addr32(&lsB[wave][1][0]) + (uint32_t)lane * 16u;

  const bool loader = (lane < 27);   // 27 lanes cover the 432B row
  const bool tail   = (lane < 12);   // bits 96..107 of the last 32-bit word

  // Running global source addresses (byte pointers), advanced by stride rows.
  const uint64_t gStep = (uint64_t)stride * ROW_BYTES;
  uint64_t gA = (uint64_t)(uintptr_t)A + (uint64_t)gwave * ROW_BYTES + (uint64_t)lane * 16u;
  uint64_t gB = (uint64_t)(uintptr_t)B + (uint64_t)gwave * ROW_BYTES + (uint64_t)lane * 16u;

  // Running output pointers.
  v4f*   srow = (v4f*)(S + (size_t)gwave * BITS) + lane;   // lane's 16B slice
  float* cptr = C + gwave;
  const size_t sStep = (size_t)stride * (BITS / 4);        // in v4f units per lane row step
  const size_t cStep = (size_t)stride;

  long r = gwave;
  if (r < (long)nrows && loader) {
    async_copy_b128(aOff[0], gA);
    async_copy_b128(bOff[0], gB);
  }

  int buf = 0;
  for (; r < (long)nrows; r += stride, buf ^= 1) {
    const long rn = r + stride;
    gA += gStep;                       // now points at row rn
    gB += gStep;
    if (rn < (long)nrows) {
      if (loader) {
        async_copy_b128(aOff[buf ^ 1], gA);
        async_copy_b128(bOff[buf ^ 1], gB);
      }
      s_wait_async_le2();              // current row's 2 DMAs done; next row's in flight
    } else {
      s_wait_async_le0();              // drain
    }

    const float* la = &lsA[wave][buf][0];
    const float* lb = &lsB[wave][buf][0];

    // wave32 bit-pack: bit (32j + lane) of the row; results are uniform SGPRs.
    uint32_t ma0 = __builtin_amdgcn_ballot_w32(la[lane]      != 0.0f);
    uint32_t ma1 = __builtin_amdgcn_ballot_w32(la[32 + lane] != 0.0f);
    uint32_t ma2 = __builtin_amdgcn_ballot_w32(la[64 + lane] != 0.0f);
    uint32_t ma3 = __builtin_amdgcn_ballot_w32(tail && (la[96 + lane] != 0.0f));
    uint32_t mb0 = __builtin_amdgcn_ballot_w32(lb[lane]      != 0.0f);
    uint32_t mb1 = __builtin_amdgcn_ballot_w32(lb[32 + lane] != 0.0f);
    uint32_t mb2 = __builtin_amdgcn_ballot_w32(lb[64 + lane] != 0.0f);
    uint32_t mb3 = __builtin_amdgcn_ballot_w32(tail && (lb[96 + lane] != 0.0f));

    uint64_t alo = (uint64_t)ma0 | ((uint64_t)ma1 << 32);
    uint64_t ahi = (uint64_t)ma2 | ((uint64_t)ma3 << 32);
    uint64_t blo = (uint64_t)mb0 | ((uint64_t)mb1 << 32);
    uint64_t bhi = (uint64_t)mb2 | ((uint64_t)mb3 << 32);

    // The entire 108-gate ripple-carry chain: one 128-bit add (scalar ALU).
    uint64_t slo = alo + blo;
    uint64_t shi = ahi + bhi + (uint64_t)(slo < alo);

    const uint32_t w0 = (uint32_t)slo;
    const uint32_t w1 = (uint32_t)(slo >> 32);
    const uint32_t w2 = (uint32_t)shi;
    const uint32_t w3 = (uint32_t)(shi >> 32);

    // Lane l owns output bits 4l..4l+3 (always within one 32-bit word).
    const int      sel  = lane >> 3;
    const uint32_t word = (sel == 0) ? w0 : (sel == 1) ? w1 : (sel == 2) ? w2 : w3;
    const uint32_t nib  = (word >> ((lane & 7) * 4)) & 0xFu;

    if (loader) {
      v4f v;
      v.x = (float)( nib       & 1u);
      v.y = (float)((nib >> 1) & 1u);
      v.z = (float)((nib >> 2) & 1u);
      v.w = (float)((nib >> 3) & 1u);
      __builtin_nontemporal_store(v, srow);        // one b128 NT store per lane
    } else if (lane == 27) {
      __builtin_nontemporal_store((float)((shi >> 44) & 1u), cptr);  // carry-out = bit 108
    }
    srow += sStep;
    cptr += cStep;
  }
}

extern "C" void kernel_launch(void* const* d_in, const int* in_sizes, int n_in,
                              void* d_out, int out_size, void* d_ws, size_t ws_size,
                              hipStream_t stream) {
  (void)n_in; (void)d_ws; (void)ws_size; (void)out_size;
  const float* A = (const float*)d_in[0];
  const float* B = (const float*)d_in[1];
  const int nrows = in_sizes[0] / BITS;

  float* S = (float*)d_out;                 // [nrows, 108]
  float* C = S + (size_t)nrows * BITS;      // [nrows, 1], concatenated after S

  const int blocks = 2048;                  // 16384 waves -> 16 rows/wave
  VecRippleCarryAdder108Bit_kernel<<<blocks, BLOCK_THREADS, 0, stream>>>(A, B, S, C, nrows);
}